// SeparateLogLikelihoodLayer_26362509263162
// MI455X (gfx1250) — compile-verified
//
#include <hip/hip_runtime.h>
#include <hip/hip_bf16.h>
#include <math.h>

typedef __attribute__((ext_vector_type(2))) float v2f;
typedef __attribute__((ext_vector_type(8))) float v8f;

#define BB 64
#define SS 1000
#define KK 12
#define NN 80
#define LSTR 81            // padded LDS row stride (gcd(81,64)=1 -> conflict-free)
#define NTILES ((SS + 15) / 16)
#define EPS 1e-5f
#define LOG_2PI_F 1.8378770664093453f

__launch_bounds__(256)
__global__ void mvn_loglik_kernel(const float* __restrict__ x,
                                  const float* __restrict__ mu,
                                  const float* __restrict__ sigma,
                                  float* __restrict__ out) {
  __shared__ float Llds[NN * LSTR];   // sigma -> Cholesky L (lower)
  __shared__ float Wlds[NN * LSTR];   // W = L^{-1}, row-major, zero upper
  __shared__ float mulds[NN];
  __shared__ float cconst_s;

  const int b   = blockIdx.x / KK;
  const int k   = blockIdx.x % KK;
  const int tid = threadIdx.x;

  // ---------------- Phase A: Cholesky + triangular inverse (per b,k) ----------
  const float* sig = sigma + (size_t)(b * KK + k) * (NN * NN);
  for (int idx = tid; idx < NN * NN; idx += 256) {
    int i = idx / NN, j = idx % NN;
    float v = sig[idx];
    if (i == j) v += EPS;
    Llds[i * LSTR + j] = v;
  }
  if (tid < NN) mulds[tid] = mu[(size_t)(b * KK + k) * NN + tid];
  __syncthreads();

  // In-place lower Cholesky, column by column.
  for (int j = 0; j < NN; ++j) {
    if (tid == 0) Llds[j * LSTR + j] = sqrtf(Llds[j * LSTR + j]);
    __syncthreads();
    float d = Llds[j * LSTR + j];
    for (int i = j + 1 + tid; i < NN; i += 256) Llds[i * LSTR + j] /= d;
    __syncthreads();
    // Trailing update: rows parallel over threads, reads only finished column j.
    for (int i = j + 1 + tid; i < NN; i += 256) {
      float lij = Llds[i * LSTR + j];
      for (int m = j + 1; m <= i; ++m)
        Llds[i * LSTR + m] -= lij * Llds[m * LSTR + j];
    }
    __syncthreads();
  }

  if (tid == 0) {
    float s = 0.0f;
    for (int i = 0; i < NN; ++i) s += logf(Llds[i * LSTR + i]);
    cconst_s = -s - 0.5f * (float)NN * LOG_2PI_F;
  }

  // W = L^{-1}: thread j computes column j by forward substitution.
  if (tid < NN) {
    const int j = tid;
    for (int i = 0; i < NN; ++i) {
      if (i < j) { Wlds[i * LSTR + j] = 0.0f; continue; }
      float s = (i == j) ? 1.0f : 0.0f;
      for (int m = j; m < i; ++m) s -= Llds[i * LSTR + m] * Wlds[m * LSTR + j];
      Wlds[i * LSTR + j] = s / Llds[i * LSTR + i];
    }
  }
  __syncthreads();
  const float cconst = cconst_s;

  // ---------------- Phase B: Z = Diff * W^T via f32 WMMA, quad = rowsum(Z^2) --
  const int wave  = tid >> 5;
  const int lane  = tid & 31;
  const int r     = lane & 15;                // row/col within 16-tile
  const int khalf = (lane >= 16) ? 2 : 0;     // K pair held by this lane half
  const float* xb = x + (size_t)b * SS * NN;
  float* outb     = out + (size_t)b * SS * KK + k;

  for (int t = wave; t < NTILES; t += 8) {
    const int s0 = t * 16;
    int srow = s0 + r;
    if (srow > SS - 1) srow = SS - 1;         // clamp loads; stores predicated
    const float* xrow = xb + (size_t)srow * NN;

    // A fragments for all 20 K-steps (16 rows x 80 K), reused for 5 N-tiles.
    v2f af[20];
#pragma unroll
    for (int kk = 0; kk < 20; ++kk) {
      const int j = kk * 4 + khalf;
      af[kk].x = xrow[j]     - mulds[j];
      af[kk].y = xrow[j + 1] - mulds[j + 1];
    }

    float acc[8] = {0.f, 0.f, 0.f, 0.f, 0.f, 0.f, 0.f, 0.f};
#pragma unroll
    for (int nt = 0; nt < 5; ++nt) {
      const int n = nt * 16 + r;              // output column = W row
      v8f c = {0.f, 0.f, 0.f, 0.f, 0.f, 0.f, 0.f, 0.f};
#pragma unroll
      for (int kk = 0; kk < 20; ++kk) {
        const int j = kk * 4 + khalf;
        v2f bf;
        bf.x = Wlds[n * LSTR + j];            // B[kj][n] = W[n][kj]
        bf.y = Wlds[n * LSTR + j + 1];
        c = __builtin_amdgcn_wmma_f32_16x16x4_f32(
                false, af[kk], false, bf, (short)0, c, false, false);
      }
#pragma unroll
      for (int g = 0; g < 8; ++g) acc[g] += c[g] * c[g];
    }

    // Reduce across each 16-lane half (C/D layout: lanes = N, halves = M 0-7/8-15).
#pragma unroll
    for (int off = 1; off <= 8; off <<= 1) {
#pragma unroll
      for (int g = 0; g < 8; ++g) acc[g] += __shfl_xor(acc[g], off, 32);
    }

    if ((lane & 15) == 0) {
      const int mbase = (lane >= 16) ? 8 : 0;
#pragma unroll
      for (int g = 0; g < 8; ++g) {
        const int s = s0 + mbase + g;
        if (s < SS) outb[(size_t)s * KK] = -0.5f * acc[g] + cconst;
      }
    }
  }
}

extern "C" void kernel_launch(void* const* d_in, const int* in_sizes, int n_in,
                              void* d_out, int out_size, void* d_ws, size_t ws_size,
                              hipStream_t stream) {
  (void)in_sizes; (void)n_in; (void)d_ws; (void)ws_size; (void)out_size;
  const float* x     = (const float*)d_in[0];
  const float* mu    = (const float*)d_in[1];
  const float* sigma = (const float*)d_in[2];
  float* out         = (float*)d_out;
  mvn_loglik_kernel<<<BB * KK, 256, 0, stream>>>(x, mu, sigma, out);
}